// interpolator_23330262352163
// MI455X (gfx1250) — compile-verified
//
#include <hip/hip_runtime.h>

// ---------------------------------------------------------------------------
// Triangle-split bilinear interpolation, B-batch, on MI455X (gfx1250).
// Memory-bound: ~100 MB HBM traffic -> ~5us floor at 23.3 TB/s.
// Strategy:
//   1) transpose x (B,N) -> xT (N,B) so one node's 8 batch values are one
//      contiguous 32B chunk (2x b128 loads per gathered node).
//   2) per-target kernel: 2 NT coord loads, weight math, 3 node gathers
//      (6x global_load_b128, temporal -> stays in 192MB L2), 8 NT stores.
// Nontemporal hints keep the 64MB output stream + 16MB coord stream from
// evicting the 8MB L2-resident field.
// ---------------------------------------------------------------------------

__global__ __launch_bounds__(256) void xpose_kernel(
    const float* __restrict__ x, float* __restrict__ xT, int N, int B)
{
    int n = blockIdx.x * blockDim.x + threadIdx.x;
    int b = blockIdx.y;
    if (n >= N) return;
    // coalesced read along n, 32B-strided write: one-time 16MB of traffic
    xT[(size_t)n * B + b] = x[(size_t)b * N + n];
}

// Shared per-target cell/weight computation, matching reference math order.
__device__ __forceinline__ void tri_weights(
    float tx, float ty, float sx0, float sy0, float dx, float dy,
    int nx, int ny,
    int& n0, int& n1, int& n2, float& w0, float& w1, float& w2)
{
    float fx = (tx - sx0) / dx;
    float fy = (ty - sy0) / dy;
    float ixf = fminf(fmaxf(floorf(fx), 0.0f), (float)(nx - 2));
    float iyf = fminf(fmaxf(floorf(fy), 0.0f), (float)(ny - 2));
    float u = fx - ixf;
    float v = fy - iyf;
    int ii = (int)ixf;
    int jj = (int)iyf;
    int base = jj * nx + ii;
    bool lower = (u + v) <= 1.0f;
    // lower triangle: nodes {base, base+1, base+nx}, w = {1-u-v, u, v}
    // upper triangle: nodes {base+nx+1, base+1, base+nx}, w = {u+v-1, 1-v, 1-u}
    n0 = lower ? base : (base + nx + 1);
    n1 = base + 1;
    n2 = base + nx;
    w0 = lower ? (1.0f - u - v) : (u + v - 1.0f);
    w1 = lower ? u : (1.0f - v);
    w2 = lower ? v : (1.0f - u);
}

// Fast path: B == 8, field pre-transposed to (N, 8).
__global__ __launch_bounds__(256) void interp_b8_kernel(
    const float* __restrict__ xT,
    const float* __restrict__ src_x, const float* __restrict__ src_y,
    const float* __restrict__ tgt_x, const float* __restrict__ tgt_y,
    const int* __restrict__ nxp, const int* __restrict__ nyp,
    float* __restrict__ out, int T, int N)
{
    int t = blockIdx.x * blockDim.x + threadIdx.x;
    if (t >= T) return;

    const int nx = *nxp;                 // uniform -> s_load
    const int ny = *nyp;
    const float sx0 = src_x[0];
    const float sx1 = src_x[N - 1];      // linspace: min at [0], max at [N-1]
    const float sy0 = src_y[0];
    const float sy1 = src_y[N - 1];
    const float dx = (sx1 - sx0) / (float)(nx - 1);
    const float dy = (sy1 - sy0) / (float)(ny - 1);

    // streamed once: non-temporal loads (TH_NT)
    float tx = __builtin_nontemporal_load(tgt_x + t);
    float ty = __builtin_nontemporal_load(tgt_y + t);

    int n0, n1, n2;
    float w0, w1, w2;
    tri_weights(tx, ty, sx0, sy0, dx, dy, nx, ny, n0, n1, n2, w0, w1, w2);

    // 3 node gathers, each 32B contiguous (8 batches) -> 2x b128 per node
    const float4* p0 = (const float4*)(xT + (size_t)n0 * 8);
    const float4* p1 = (const float4*)(xT + (size_t)n1 * 8);
    const float4* p2 = (const float4*)(xT + (size_t)n2 * 8);
    float4 a0 = p0[0], a1 = p0[1];
    float4 b0 = p1[0], b1 = p1[1];
    float4 c0 = p2[0], c1 = p2[1];

    float r0 = fmaf(w0, a0.x, fmaf(w1, b0.x, w2 * c0.x));
    float r1 = fmaf(w0, a0.y, fmaf(w1, b0.y, w2 * c0.y));
    float r2 = fmaf(w0, a0.z, fmaf(w1, b0.z, w2 * c0.z));
    float r3 = fmaf(w0, a0.w, fmaf(w1, b0.w, w2 * c0.w));
    float r4 = fmaf(w0, a1.x, fmaf(w1, b1.x, w2 * c1.x));
    float r5 = fmaf(w0, a1.y, fmaf(w1, b1.y, w2 * c1.y));
    float r6 = fmaf(w0, a1.z, fmaf(w1, b1.z, w2 * c1.z));
    float r7 = fmaf(w0, a1.w, fmaf(w1, b1.w, w2 * c1.w));

    // output written once, never re-read: non-temporal stores, coalesced per b
    __builtin_nontemporal_store(r0, out + 0 * (size_t)T + t);
    __builtin_nontemporal_store(r1, out + 1 * (size_t)T + t);
    __builtin_nontemporal_store(r2, out + 2 * (size_t)T + t);
    __builtin_nontemporal_store(r3, out + 3 * (size_t)T + t);
    __builtin_nontemporal_store(r4, out + 4 * (size_t)T + t);
    __builtin_nontemporal_store(r5, out + 5 * (size_t)T + t);
    __builtin_nontemporal_store(r6, out + 6 * (size_t)T + t);
    __builtin_nontemporal_store(r7, out + 7 * (size_t)T + t);
}

// Generic fallback: any B, native (B, N) layout, no workspace needed.
__global__ __launch_bounds__(256) void interp_generic_kernel(
    const float* __restrict__ x,
    const float* __restrict__ src_x, const float* __restrict__ src_y,
    const float* __restrict__ tgt_x, const float* __restrict__ tgt_y,
    const int* __restrict__ nxp, const int* __restrict__ nyp,
    float* __restrict__ out, int T, int N, int B)
{
    int t = blockIdx.x * blockDim.x + threadIdx.x;
    if (t >= T) return;

    const int nx = *nxp;
    const int ny = *nyp;
    const float sx0 = src_x[0];
    const float sx1 = src_x[N - 1];
    const float sy0 = src_y[0];
    const float sy1 = src_y[N - 1];
    const float dx = (sx1 - sx0) / (float)(nx - 1);
    const float dy = (sy1 - sy0) / (float)(ny - 1);

    float tx = __builtin_nontemporal_load(tgt_x + t);
    float ty = __builtin_nontemporal_load(tgt_y + t);

    int n0, n1, n2;
    float w0, w1, w2;
    tri_weights(tx, ty, sx0, sy0, dx, dy, nx, ny, n0, n1, n2, w0, w1, w2);

    for (int b = 0; b < B; ++b) {
        const float* xb = x + (size_t)b * N;
        float r = fmaf(w0, xb[n0], fmaf(w1, xb[n1], w2 * xb[n2]));
        __builtin_nontemporal_store(r, out + (size_t)b * T + t);
    }
}

extern "C" void kernel_launch(void* const* d_in, const int* in_sizes, int n_in,
                              void* d_out, int out_size, void* d_ws, size_t ws_size,
                              hipStream_t stream) {
    const float* x     = (const float*)d_in[0];   // (B, N)
    const float* src_x = (const float*)d_in[1];   // (N)
    const float* src_y = (const float*)d_in[2];   // (N)
    const float* tgt_x = (const float*)d_in[3];   // (T)
    const float* tgt_y = (const float*)d_in[4];   // (T)
    const int*   nxp   = (const int*)d_in[5];     // scalar on device
    const int*   nyp   = (const int*)d_in[6];     // scalar on device
    float* out = (float*)d_out;                   // (B, T)

    const int N = in_sizes[1];
    const int B = in_sizes[0] / N;
    const int T = in_sizes[3];

    const int threads = 256;
    const int tgrid = (T + threads - 1) / threads;

    if (B == 8 && ws_size >= (size_t)N * 8 * sizeof(float)) {
        float* xT = (float*)d_ws;                 // (N, 8)
        dim3 xg((N + threads - 1) / threads, 8);
        xpose_kernel<<<xg, threads, 0, stream>>>(x, xT, N, 8);
        interp_b8_kernel<<<tgrid, threads, 0, stream>>>(
            xT, src_x, src_y, tgt_x, tgt_y, nxp, nyp, out, T, N);
    } else {
        interp_generic_kernel<<<tgrid, threads, 0, stream>>>(
            x, src_x, src_y, tgt_x, tgt_y, nxp, nyp, out, T, N, B);
    }
}